// DGLGATModel_77927886619026
// MI455X (gfx1250) — compile-verified
//
#include <hip/hip_runtime.h>
#include <hip/hip_bf16.h>
#include <math.h>

typedef __attribute__((ext_vector_type(2))) float v2f;
typedef __attribute__((ext_vector_type(8))) float v8f;

#define N_NODES 50000
#define N_EDGES 800000
#define HEADS   8
#define FEAT    32
#define NCOL    256   // HEADS*FEAT

// ---------------------------------------------------------------------------
// CSR construction (dst-indexed), built once per launch, reused by both layers
// ---------------------------------------------------------------------------
__global__ void k_zero_i32(int* __restrict__ p, int n) {
  int i = blockIdx.x * blockDim.x + threadIdx.x;
  if (i < n) p[i] = 0;
}

__global__ void k_count(const int* __restrict__ dst, int* __restrict__ cnt, int E) {
  int i = blockIdx.x * blockDim.x + threadIdx.x;
  if (i < E) atomicAdd(&cnt[dst[i]], 1);
}

// single-block exclusive scan over 50k counts -> rowptr[N+1], cursor copy
__global__ void k_exscan(const int* __restrict__ cnt, int* __restrict__ rowptr,
                         int* __restrict__ cursor, int n) {
  __shared__ int sums[1024];
  int t = threadIdx.x;
  int chunk = (n + 1023) >> 10;
  int s0 = t * chunk; if (s0 > n) s0 = n;
  int s1 = s0 + chunk; if (s1 > n) s1 = n;
  int sum = 0;
  for (int i = s0; i < s1; ++i) sum += cnt[i];
  sums[t] = sum;
  __syncthreads();
  for (int off = 1; off < 1024; off <<= 1) {
    int v = (t >= off) ? sums[t - off] : 0;
    __syncthreads();
    sums[t] += v;
    __syncthreads();
  }
  int run = (t == 0) ? 0 : sums[t - 1];
  for (int i = s0; i < s1; ++i) {
    int c = cnt[i];
    rowptr[i] = run;
    cursor[i] = run;
    run += c;
  }
  if (t == 0) rowptr[n] = sums[1023];
}

__global__ void k_scatter(const int* __restrict__ dst, int* __restrict__ cursor,
                          int* __restrict__ eidx, int E) {
  int i = blockIdx.x * blockDim.x + threadIdx.x;
  if (i < E) {
    int p = atomicAdd(&cursor[dst[i]], 1);
    eidx[p] = i;
  }
}

// ---------------------------------------------------------------------------
// Dense projection h = A[M,K] @ B[K,256] via V_WMMA_F32_16X16X4_F32.
// One wave -> 16x64 output strip (4 N-tiles share one A fragment).
// A frag: lanes 0-15 hold K=koff..koff+1 of row (lane&15), koff = (lane>=16)?2:0
// B frag: VGPR0 = row K=koff, VGPR1 = row K=koff+1, col = lane&15
// D: VGPR j -> row m0 + j + ((lane>=16)?8:0), col = lane&15
// ---------------------------------------------------------------------------
__global__ void k_gemm_wmma(const float* __restrict__ A, const float* __restrict__ B,
                            float* __restrict__ C, int M, int K) {
  int wid  = (blockIdx.x * blockDim.x + threadIdx.x) >> 5;
  int lane = threadIdx.x & 31;
  int mtiles = M >> 4;
  if (wid >= mtiles * 4) return;
  int mt = wid >> 2, ng = wid & 3;
  int m0 = mt << 4, n0 = ng << 6;
  int rc   = lane & 15;              // row (for A) / col (for B,C,D)
  int koff = (lane >> 4) << 1;       // 0 or 2
  int hi8  = (lane >> 4) << 3;       // 0 or 8

  v8f acc0 = {}, acc1 = {}, acc2 = {}, acc3 = {};
  const float* arow = A + (m0 + rc) * K;

  #pragma unroll 4
  for (int k = 0; k < K; k += 4) {
    v2f a;
    a.x = arow[k + koff];
    a.y = arow[k + koff + 1];
    const float* bp = B + (k + koff) * NCOL + n0 + rc;
    v2f b;
    b.x = bp[0];       b.y = bp[NCOL];
    acc0 = __builtin_amdgcn_wmma_f32_16x16x4_f32(false, a, false, b, (short)0, acc0, false, false);
    b.x = bp[16];      b.y = bp[NCOL + 16];
    acc1 = __builtin_amdgcn_wmma_f32_16x16x4_f32(false, a, false, b, (short)0, acc1, false, false);
    b.x = bp[32];      b.y = bp[NCOL + 32];
    acc2 = __builtin_amdgcn_wmma_f32_16x16x4_f32(false, a, false, b, (short)0, acc2, false, false);
    b.x = bp[48];      b.y = bp[NCOL + 48];
    acc3 = __builtin_amdgcn_wmma_f32_16x16x4_f32(false, a, false, b, (short)0, acc3, false, false);
  }

  float* crow = C + (m0 + hi8) * NCOL + n0 + rc;
  #pragma unroll
  for (int j = 0; j < 8; ++j) {
    crow[j * NCOL +  0] = acc0[j];
    crow[j * NCOL + 16] = acc1[j];
    crow[j * NCOL + 32] = acc2[j];
    crow[j * NCOL + 48] = acc3[j];
  }
}

// ---------------------------------------------------------------------------
// el[n,h] = <h[n,h,:], attn_l[h,:]> ; er likewise. One thread per (n,h).
// ---------------------------------------------------------------------------
__global__ void k_attn(const float* __restrict__ h, const float* __restrict__ al,
                       const float* __restrict__ ar, float* __restrict__ el,
                       float* __restrict__ er) {
  int idx = blockIdx.x * blockDim.x + threadIdx.x;
  if (idx >= N_NODES * HEADS) return;
  int hd = idx & (HEADS - 1);
  int n  = idx >> 3;
  const float* hp  = h + n * NCOL + hd * FEAT;
  const float* alp = al + hd * FEAT;
  const float* arp = ar + hd * FEAT;
  float sl = 0.f, sr = 0.f;
  #pragma unroll
  for (int f = 0; f < FEAT; ++f) {
    float v = hp[f];
    sl = fmaf(v, alp[f], sl);
    sr = fmaf(v, arp[f], sr);
  }
  el[idx] = sl;
  er[idx] = sr;
}

// ---------------------------------------------------------------------------
// Gather aggregation with one-pass online softmax (flash-style).
// One wave32 per (dst node, head); lane == feature index.
// out[n,h,f] = sum_e softmax_e(leakyrelu(el[src]+er[n])) * h[src,h,f] + bias
// ---------------------------------------------------------------------------
__global__ void k_aggregate(const float* __restrict__ h, const float* __restrict__ el,
                            const float* __restrict__ er, const float* __restrict__ bias,
                            const int* __restrict__ rowptr, const int* __restrict__ eidx,
                            const int* __restrict__ src, float* __restrict__ out) {
  int wid  = (blockIdx.x * blockDim.x + threadIdx.x) >> 5;
  int lane = threadIdx.x & 31;
  if (wid >= N_NODES * HEADS) return;
  int hd = wid & (HEADS - 1);
  int n  = wid >> 3;
  int s0 = rowptr[n], s1 = rowptr[n + 1];
  float ern = er[wid];

  float m = -INFINITY, ssum = 0.f, acc = 0.f;
  for (int j = s0; j < s1; ++j) {
    int e  = eidx[j];
    int sn = src[e];
    float ev = el[sn * HEADS + hd] + ern;
    ev = (ev > 0.f) ? ev : 0.2f * ev;                  // leaky_relu(0.2)
    float mn = fmaxf(m, ev);
    float sc = __expf(m - mn);                          // 0 on first iter (m=-inf)
    float p  = __expf(ev - mn);
    ssum = ssum * sc + p;
    acc  = acc  * sc + p * h[sn * NCOL + hd * FEAT + lane];
    m = mn;
  }
  float r = (s1 > s0) ? (acc / ssum) : 0.f;             // isolated node -> bias only
  out[n * NCOL + hd * FEAT + lane] = r + bias[hd * FEAT + lane];
}

// ---------------------------------------------------------------------------
// h1act[n,f] = mean_h elu(agg[n,h,f])   (dropout is identity in eval)
// ---------------------------------------------------------------------------
__global__ void k_elu_headmean(const float* __restrict__ agg, float* __restrict__ out) {
  int idx = blockIdx.x * blockDim.x + threadIdx.x;
  if (idx >= N_NODES * FEAT) return;
  int f = idx & (FEAT - 1);
  int n = idx >> 5;
  float s = 0.f;
  #pragma unroll
  for (int hd = 0; hd < HEADS; ++hd) {
    float v = agg[n * NCOL + hd * FEAT + f];
    s += (v > 0.f) ? v : (__expf(v) - 1.f);
  }
  out[idx] = s * (1.f / HEADS);
}

// ---------------------------------------------------------------------------
extern "C" void kernel_launch(void* const* d_in, const int* in_sizes, int n_in,
                              void* d_out, int out_size, void* d_ws, size_t ws_size,
                              hipStream_t stream) {
  const float* x   = (const float*)d_in[0];
  const int*   src = (const int*)  d_in[1];
  const int*   dst = (const int*)  d_in[2];
  const float* W1  = (const float*)d_in[3];
  const float* al1 = (const float*)d_in[4];
  const float* ar1 = (const float*)d_in[5];
  const float* b1  = (const float*)d_in[6];
  const float* W2  = (const float*)d_in[7];
  const float* al2 = (const float*)d_in[8];
  const float* ar2 = (const float*)d_in[9];
  const float* b2  = (const float*)d_in[10];

  float* wsf = (float*)d_ws;
  float* h_buf = wsf;                                    // 50000*256
  float* agg1  = h_buf + (size_t)N_NODES * NCOL;         // 50000*256
  float* elb   = agg1  + (size_t)N_NODES * NCOL;         // 50000*8
  float* erb   = elb   + (size_t)N_NODES * HEADS;        // 50000*8
  float* h1act = erb   + (size_t)N_NODES * HEADS;        // 50000*32
  int* ibase   = (int*)(h1act + (size_t)N_NODES * FEAT);
  int* rowptr  = ibase;                                  // N+1
  int* cursor  = rowptr + (N_NODES + 2);                 // N
  int* cnts    = cursor + N_NODES;                       // N
  int* eidx    = cnts   + N_NODES;                       // E

  // --- CSR build (graph shared by both layers) ---
  k_zero_i32<<<(N_NODES + 255) / 256, 256, 0, stream>>>(cnts, N_NODES);
  k_count   <<<(N_EDGES + 255) / 256, 256, 0, stream>>>(dst, cnts, N_EDGES);
  k_exscan  <<<1, 1024, 0, stream>>>(cnts, rowptr, cursor, N_NODES);
  k_scatter <<<(N_EDGES + 255) / 256, 256, 0, stream>>>(dst, cursor, eidx, N_EDGES);

  const int gemm_blocks = ((N_NODES / 16) * 4 * 32 + 255) / 256;   // 1563
  const int nh_blocks   = (N_NODES * HEADS + 255) / 256;           // 1563
  const int agg_blocks  = (N_NODES * HEADS * 32) / 256;            // 50000
  const int nf_blocks   = (N_NODES * FEAT + 255) / 256;            // 6250

  // --- Layer 1 ---
  k_gemm_wmma <<<gemm_blocks, 256, 0, stream>>>(x, W1, h_buf, N_NODES, 128);
  k_attn      <<<nh_blocks,  256, 0, stream>>>(h_buf, al1, ar1, elb, erb);
  k_aggregate <<<agg_blocks, 256, 0, stream>>>(h_buf, elb, erb, b1, rowptr, eidx, src, agg1);
  k_elu_headmean<<<nf_blocks, 256, 0, stream>>>(agg1, h1act);

  // --- Layer 2 (writes final output directly) ---
  k_gemm_wmma <<<gemm_blocks, 256, 0, stream>>>(h1act, W2, h_buf, N_NODES, 32);
  k_attn      <<<nh_blocks,  256, 0, stream>>>(h_buf, al2, ar2, elb, erb);
  k_aggregate <<<agg_blocks, 256, 0, stream>>>(h_buf, elb, erb, b2, rowptr, eidx, src, (float*)d_out);
}